// GraphConvLayer_52183852646405
// MI455X (gfx1250) — compile-verified
//
#include <hip/hip_runtime.h>

typedef __attribute__((ext_vector_type(2))) float v2f;
typedef __attribute__((ext_vector_type(8))) float v8f;

#define D_IN   128
#define D_OUT  128
#define K_DIM  256               // 2 * D_IN
#define TILE_M 64                // rows per block
#define MT     4                 // 16-row M-subtiles per wave (B-register reuse x4)
#define LDS_STRIDE (K_DIM + 8)   // 264 floats: keeps 16B alignment, staggers banks

// ---------------------------------------------------------------------------
// Kernel 1: zero the aggregation buffer (must run every call: graph replays).
// ---------------------------------------------------------------------------
__global__ __launch_bounds__(256) void zero_kernel(float* __restrict__ p, int n4) {
  int i = blockIdx.x * blockDim.x + threadIdx.x;
  if (i < n4) ((float4*)p)[i] = make_float4(0.f, 0.f, 0.f, 0.f);
}

// ---------------------------------------------------------------------------
// Kernel 2: edge scatter.  agg[dst] += feat[src] * affine.
// dst is sorted -> accumulate runs of equal dst in registers, flush with
// fp32 global atomics only at run boundaries (and chunk edges).
// One wave per chunk; lane L owns floats [4L, 4L+4) of the 128-float row.
// ---------------------------------------------------------------------------
__global__ __launch_bounds__(256) void scatter_kernel(
    const float* __restrict__ feat, const float* __restrict__ affine,
    const int* __restrict__ src, const int* __restrict__ dst,
    float* __restrict__ agg, int E, int epw)
{
  const int wave = (blockIdx.x * blockDim.x + threadIdx.x) >> 5;
  const int lane = threadIdx.x & 31;
  long e = (long)wave * epw;
  if (e >= E) return;
  long eend = e + epw;
  if (eend > E) eend = E;

  float4 acc = make_float4(0.f, 0.f, 0.f, 0.f);
  int cur = __builtin_amdgcn_readfirstlane(dst[e]);

  for (; e < eend; ++e) {
    const int d = __builtin_amdgcn_readfirstlane(dst[e]);
    if (d != cur) {
      float* p = agg + (size_t)cur * D_IN + lane * 4;
      __hip_atomic_fetch_add(p + 0, acc.x, __ATOMIC_RELAXED, __HIP_MEMORY_SCOPE_AGENT);
      __hip_atomic_fetch_add(p + 1, acc.y, __ATOMIC_RELAXED, __HIP_MEMORY_SCOPE_AGENT);
      __hip_atomic_fetch_add(p + 2, acc.z, __ATOMIC_RELAXED, __HIP_MEMORY_SCOPE_AGENT);
      __hip_atomic_fetch_add(p + 3, acc.w, __ATOMIC_RELAXED, __HIP_MEMORY_SCOPE_AGENT);
      acc = make_float4(0.f, 0.f, 0.f, 0.f);
      cur = d;
    }
    const int   s = src[e];
    const float a = affine[e];
    const float4 f = ((const float4*)(feat + (size_t)s * D_IN))[lane];
    acc.x = fmaf(a, f.x, acc.x);
    acc.y = fmaf(a, f.y, acc.y);
    acc.z = fmaf(a, f.z, acc.z);
    acc.w = fmaf(a, f.w, acc.w);
  }
  float* p = agg + (size_t)cur * D_IN + lane * 4;
  __hip_atomic_fetch_add(p + 0, acc.x, __ATOMIC_RELAXED, __HIP_MEMORY_SCOPE_AGENT);
  __hip_atomic_fetch_add(p + 1, acc.y, __ATOMIC_RELAXED, __HIP_MEMORY_SCOPE_AGENT);
  __hip_atomic_fetch_add(p + 2, acc.z, __ATOMIC_RELAXED, __HIP_MEMORY_SCOPE_AGENT);
  __hip_atomic_fetch_add(p + 3, acc.w, __ATOMIC_RELAXED, __HIP_MEMORY_SCOPE_AGENT);
}

// ---------------------------------------------------------------------------
// Kernel 3: out = [feat | agg] @ W + b via V_WMMA_F32_16X16X4_F32.
// Block = 256 threads (8 waves). Block owns 64 output rows; wave w owns
// output columns [16w, 16w+16) and 4 M-subtiles that share B registers.
// K = 256, stepped by 4 per WMMA -> 4 WMMAs per B load pair.
// ---------------------------------------------------------------------------
__global__ __launch_bounds__(256) void gemm_kernel(
    const float* __restrict__ feat, const float* __restrict__ agg,
    const float* __restrict__ W, const float* __restrict__ bias,
    float* __restrict__ out, int nrows)
{
  __shared__ float As[TILE_M * LDS_STRIDE];   // 64 x 264 floats = 66 KB

  const int tid    = threadIdx.x;
  const int waveId = tid >> 5;
  const int lane   = tid & 31;
  const int row0   = blockIdx.x * TILE_M;

  // Stage the 64x256 A tile into LDS: cols [0,128) = feat, [128,256) = agg.
  // 4096 float4 transfers, 16 per thread, fully coalesced (512B per row half).
#pragma unroll
  for (int i = 0; i < 16; ++i) {
    int q  = tid + i * 256;   // float4 id 0..4095
    int r  = q >> 6;          // row 0..63
    int c4 = q & 63;          // float4 col 0..63
    int rr = row0 + r;
    if (rr >= nrows) rr = nrows - 1;   // clamp: only last (partial) block
    float4 v;
    if (c4 < 32) v = ((const float4*)(feat + (size_t)rr * D_IN))[c4];
    else         v = ((const float4*)(agg  + (size_t)rr * D_IN))[c4 - 32];
    *(float4*)&As[r * LDS_STRIDE + c4 * 4] = v;
  }
  __syncthreads();

  const int ncol = waveId * 16 + (lane & 15);
  const int m    = lane & 15;          // A: M = lane % 16 within a subtile
  const int kb   = (lane >> 4) << 1;   // A/B: K sub-block 0 or 2

  // C/D layout: VGPR r -> M = r (lanes 0-15) or r+8 (lanes 16-31), N = lane%16.
  v8f acc[MT];
  const float bv = bias[ncol];
#pragma unroll
  for (int t = 0; t < MT; ++t)
#pragma unroll
    for (int r = 0; r < 8; ++r) acc[t][r] = bv;

#pragma unroll 2
  for (int k = 0; k < K_DIM; k += 4) {
    // B operand (shared by all 4 M-subtiles): W[k+kb][ncol], W[k+kb+1][ncol]
    v2f bm;
    bm.x = W[(size_t)(k + kb)     * D_OUT + ncol];
    bm.y = W[(size_t)(k + kb + 1) * D_OUT + ncol];
#pragma unroll
    for (int t = 0; t < MT; ++t) {
      const v2f a = *(const v2f*)&As[(t * 16 + m) * LDS_STRIDE + k + kb];
      acc[t] = __builtin_amdgcn_wmma_f32_16x16x4_f32(
          /*neg_a=*/false, a, /*neg_b=*/false, bm,
          /*c_mod=*/(short)0, acc[t], /*reuse_a=*/false, /*reuse_b=*/false);
    }
  }

  const int mrow = (lane >> 4) << 3;   // 0 or 8
  if (row0 + TILE_M <= nrows) {
    // Fast path: full tile, no per-row guards.
#pragma unroll
    for (int t = 0; t < MT; ++t)
#pragma unroll
      for (int r = 0; r < 8; ++r)
        out[(size_t)(row0 + t * 16 + mrow + r) * D_OUT + ncol] = acc[t][r];
  } else {
#pragma unroll
    for (int t = 0; t < MT; ++t)
#pragma unroll
      for (int r = 0; r < 8; ++r) {
        const int row = row0 + t * 16 + mrow + r;
        if (row < nrows) out[(size_t)row * D_OUT + ncol] = acc[t][r];
      }
  }
}

// ---------------------------------------------------------------------------
extern "C" void kernel_launch(void* const* d_in, const int* in_sizes, int n_in,
                              void* d_out, int out_size, void* d_ws, size_t ws_size,
                              hipStream_t stream) {
  const float* feat   = (const float*)d_in[0];
  const float* affine = (const float*)d_in[1];
  const int*   src    = (const int*)d_in[2];
  const int*   dst    = (const int*)d_in[3];
  const float* W      = (const float*)d_in[4];
  const float* bias   = (const float*)d_in[5];
  float*       out    = (float*)d_out;
  float*       agg    = (float*)d_ws;   // N * 128 floats = 25.6 MB scratch

  const int N = in_sizes[0] / D_IN;
  const int E = in_sizes[1];

  // 1) zero agg
  const int n4 = (N * D_IN) / 4;
  zero_kernel<<<(n4 + 255) / 256, 256, 0, stream>>>(agg, n4);

  // 2) edge scatter (4096 waves; ~153 edges per wave, run-length accumulated)
  const int nwaves = 4096;
  const int epw = (E + nwaves - 1) / nwaves;
  scatter_kernel<<<nwaves / 8, 256, 0, stream>>>(feat, affine, src, dst, agg, E, epw);

  // 3) fused concat + GEMM + bias via fp32 WMMA (B registers reused x4)
  gemm_kernel<<<(N + TILE_M - 1) / TILE_M, 256, 0, stream>>>(feat, agg, W, bias, out, N);
}